// model_own_32109175505028
// MI455X (gfx1250) — compile-verified
//
#include <hip/hip_runtime.h>
#include <hip/hip_bf16.h>
#include <cstddef>

typedef float v2f __attribute__((ext_vector_type(2)));
typedef float v8f __attribute__((ext_vector_type(8)));

#define KNN_K 20
#define NPT   4096
#define NB    8
#define SLOPE 0.2f
#define BN_CNT 655360.0f   // B*N*K = 8*4096*20

__device__ __forceinline__ float lrelu(float v) { return v >= 0.0f ? v : SLOPE * v; }

// ---------------------------------------------------------------------------
// transpose (3,8,4096) -> (8,3,4096)
// ---------------------------------------------------------------------------
__global__ void transpose_x_kernel(const float* __restrict__ xin, float* __restrict__ xt) {
    int i = blockIdx.x * 256 + threadIdx.x;
    if (i < 3 * NB * NPT) {
        int n = i % NPT;
        int b = (i / NPT) % NB;
        int c = i / (NPT * NB);
        xt[((size_t)b * 3 + c) * NPT + n] = xin[i];
    }
}

// ---------------------------------------------------------------------------
// per-point squared norm: xx[b][n] = sum_c feat[b][c][n]^2
// ---------------------------------------------------------------------------
template <int C>
__global__ void xx_kernel(const float* __restrict__ feat, float* __restrict__ xx) {
    int i = blockIdx.x * 256 + threadIdx.x;
    if (i < NB * NPT) {
        int b = i / NPT, n = i % NPT;
        const float* fb = feat + (size_t)b * C * NPT + n;
        float s = 0.f;
#pragma unroll
        for (int c = 0; c < C; ++c) { float v = fb[(size_t)c * NPT]; s += v * v; }
        xx[i] = s;
    }
}

// ---------------------------------------------------------------------------
// KNN: top-20 of score s_j = 2*inner(i,j) - xx[j]  (same ordering as pd row).
// Gram tiles via V_WMMA_F32_16X16X4_F32. 128 threads = 4 waves.
// Workgroup handles 16 rows (I..I+15) of one batch, scanning columns in
// 512-wide chunks staged in LDS.
// ---------------------------------------------------------------------------
template <int C>
__global__ __launch_bounds__(128) void knn_kernel(const float* __restrict__ feat,
                                                  const float* __restrict__ xx,
                                                  int* __restrict__ idx_out) {
    constexpr int KSTEPS = (C + 3) / 4;
    const int b    = blockIdx.y;
    const int I    = blockIdx.x * 16;
    const int tid  = threadIdx.x;
    const int lane = tid & 31;
    const int wave = tid >> 5;      // 0..3
    const int l16  = lane & 15;
    const int hi   = lane >> 4;     // 0 or 1

    __shared__ float sc[16 * 520];      // 16 rows x 512 cols (pad 8)
    __shared__ float mv[128 * KNN_K];   // per-thread candidate values
    __shared__ int   mi[128 * KNN_K];   // per-thread candidate indices

    const float* fb  = feat + (size_t)b * C * NPT;
    const float* xxb = xx   + (size_t)b * NPT;

    // A fragments (rows I..I+15), ISA 16x4 f32 layout:
    // lanes 0-15: K=0,1 ; lanes 16-31: K=2,3  (per 4-K step)
    v2f afrag[KSTEPS];
#pragma unroll
    for (int s = 0; s < KSTEPS; ++s) {
        const int k0 = s * 4 + hi * 2;
        float ax = (k0     < C) ? fb[(size_t)k0       * NPT + I + l16] : 0.0f;
        float ay = (k0 + 1 < C) ? fb[(size_t)(k0 + 1) * NPT + I + l16] : 0.0f;
        v2f a; a.x = ax; a.y = ay;
        afrag[s] = a;
    }

    // register-resident sorted top-20 (desc value, asc index on ties)
    float bv[KNN_K]; int bix[KNN_K];
#pragma unroll
    for (int q = 0; q < KNN_K; ++q) { bv[q] = -3.0e38f; bix[q] = 0x7fffffff; }

    const int row = tid >> 3;   // 16 rows
    const int sub = tid & 7;    // 8 scanners per row

    for (int chunk = 0; chunk < NPT; chunk += 512) {
        // 32 column tiles per chunk, 8 per wave
        for (int t = wave; t < 32; t += 4) {
            const int J = chunk + t * 16;
            v8f acc = {};
#pragma unroll
            for (int s = 0; s < KSTEPS; ++s) {
                const int k0 = s * 4 + hi * 2;
                float bx = (k0     < C) ? fb[(size_t)k0       * NPT + J + l16] : 0.0f;
                float by = (k0 + 1 < C) ? fb[(size_t)(k0 + 1) * NPT + J + l16] : 0.0f;
                v2f bf; bf.x = bx; bf.y = by;
                acc = __builtin_amdgcn_wmma_f32_16x16x4_f32(
                        false, afrag[s], false, bf, (short)0, acc, false, false);
            }
            const float xj   = xxb[J + l16];
            const int   colc = (J - chunk) + l16;
#pragma unroll
            for (int r = 0; r < 8; ++r) {
                const int m = r + hi * 8;            // C/D layout: M = r + 8*hi
                sc[m * 520 + colc] = 2.0f * acc[r] - xj;
            }
        }
        __syncthreads();

        // streaming selection over this chunk
        for (int cidx = sub; cidx < 512; cidx += 8) {
            float cv = sc[row * 520 + cidx];
            int   ci = chunk + cidx;
            if (cv > bv[KNN_K - 1] || (cv == bv[KNN_K - 1] && ci < bix[KNN_K - 1])) {
#pragma unroll
                for (int q = 0; q < KNN_K; ++q) {
                    bool better = (cv > bv[q]) || (cv == bv[q] && ci < bix[q]);
                    float tv = bv[q]; int ti = bix[q];
                    if (better) { bv[q] = cv; bix[q] = ci; cv = tv; ci = ti; }
                }
            }
        }
        __syncthreads();
    }

#pragma unroll
    for (int q = 0; q < KNN_K; ++q) { mv[tid * KNN_K + q] = bv[q]; mi[tid * KNN_K + q] = bix[q]; }
    __syncthreads();

    // merge 8 sorted lists per row (threads 0..15, one per row)
    if (tid < 16) {
        int head[8];
#pragma unroll
        for (int s = 0; s < 8; ++s) head[s] = 0;
        const int base = tid * 8;
        int* orow = idx_out + ((size_t)b * NPT + (I + tid)) * KNN_K;
        for (int q = 0; q < KNN_K; ++q) {
            float bestv = -3.0e38f; int besti = 0x7fffffff; int bests = 0;
            for (int s = 0; s < 8; ++s) {
                const int h = head[s];
                if (h < KNN_K) {
                    float v  = mv[(base + s) * KNN_K + h];
                    int   ii = mi[(base + s) * KNN_K + h];
                    if (v > bestv || (v == bestv && ii < besti)) { bestv = v; besti = ii; bests = s; }
                }
            }
            head[bests]++;
            orow[q] = besti;
        }
    }
}

// ---------------------------------------------------------------------------
// edge conv: e = [nbr-ctr, ctr] (2*CIN), y = W e (COUT x 2CIN).
// Stores pre-BN max over k and deterministic per-block sum/sumsq partials.
// ---------------------------------------------------------------------------
template <int CIN, int COUT>
__global__ __launch_bounds__(256) void edgeconv_kernel(const float* __restrict__ feat,
                                                       const int* __restrict__ idx,
                                                       const float* __restrict__ w,
                                                       float* __restrict__ pre,
                                                       float* __restrict__ part) {
    const int b = blockIdx.y, tid = threadIdx.x;
    const int n = blockIdx.x * 256 + tid;
    const float* fb = feat + (size_t)b * CIN * NPT;

    float ctr[CIN];
#pragma unroll
    for (int c = 0; c < CIN; ++c) ctr[c] = fb[(size_t)c * NPT + n];
    const int* ids = idx + ((size_t)b * NPT + n) * KNN_K;

    float ymax[COUT], s1[COUT], s2[COUT];
#pragma unroll
    for (int o = 0; o < COUT; ++o) { ymax[o] = -3.0e38f; s1[o] = 0.f; s2[o] = 0.f; }

#pragma unroll 1
    for (int k = 0; k < KNN_K; ++k) {
        const int j = ids[k];
        float e[2 * CIN];
#pragma unroll
        for (int c = 0; c < CIN; ++c) { e[c] = fb[(size_t)c * NPT + j] - ctr[c]; e[CIN + c] = ctr[c]; }
#pragma unroll
        for (int o = 0; o < COUT; ++o) {
            float y = 0.f;
#pragma unroll
            for (int c = 0; c < 2 * CIN; ++c) y += w[o * 2 * CIN + c] * e[c];
            ymax[o] = fmaxf(ymax[o], y);
            s1[o] += y; s2[o] += y * y;
        }
    }
#pragma unroll
    for (int o = 0; o < COUT; ++o) pre[((size_t)b * COUT + o) * NPT + n] = ymax[o];

    __shared__ float red[8 * 64];
    const int lane = tid & 31, wv = tid >> 5;
    if (lane == 0) for (int z = 0; z < 64; ++z) red[wv * 64 + z] = 0.f;
#pragma unroll
    for (int o = 0; o < COUT; ++o) {
        float a = s1[o], q = s2[o];
        for (int off = 16; off > 0; off >>= 1) { a += __shfl_down(a, off); q += __shfl_down(q, off); }
        if (lane == 0) { red[wv * 64 + o] = a; red[wv * 64 + 32 + o] = q; }
    }
    __syncthreads();
    if (tid < 64) {
        float s = 0.f;
        for (int w2 = 0; w2 < 8; ++w2) s += red[w2 * 64 + tid];
        part[(size_t)(blockIdx.y * gridDim.x + blockIdx.x) * 64 + tid] = s;
    }
}

// ---------------------------------------------------------------------------
// conv2: e = [gf3(nbr1-ctr3, ctr3), gf16(nbr2-ctr16, ctr16)]  (38), W 32x38
// ---------------------------------------------------------------------------
__global__ __launch_bounds__(256) void conv2_kernel(const float* __restrict__ xt,
                                                    const int* __restrict__ idx1,
                                                    const float* __restrict__ x1,
                                                    const int* __restrict__ idx2,
                                                    const float* __restrict__ w,
                                                    float* __restrict__ pre,
                                                    float* __restrict__ part) {
    const int b = blockIdx.y, tid = threadIdx.x;
    const int n = blockIdx.x * 256 + tid;
    const float* f3  = xt + (size_t)b * 3  * NPT;
    const float* f16 = x1 + (size_t)b * 16 * NPT;

    float c3[3], c16[16];
#pragma unroll
    for (int c = 0; c < 3; ++c)  c3[c]  = f3[(size_t)c * NPT + n];
#pragma unroll
    for (int c = 0; c < 16; ++c) c16[c] = f16[(size_t)c * NPT + n];
    const int* i1 = idx1 + ((size_t)b * NPT + n) * KNN_K;
    const int* i2 = idx2 + ((size_t)b * NPT + n) * KNN_K;

    float ymax[32], s1[32], s2[32];
#pragma unroll
    for (int o = 0; o < 32; ++o) { ymax[o] = -3.0e38f; s1[o] = 0.f; s2[o] = 0.f; }

#pragma unroll 1
    for (int k = 0; k < KNN_K; ++k) {
        const int j1 = i1[k], j2 = i2[k];
        float e[38];
#pragma unroll
        for (int c = 0; c < 3; ++c)  { e[c]     = f3[(size_t)c * NPT + j1] - c3[c];  e[3 + c]  = c3[c]; }
#pragma unroll
        for (int c = 0; c < 16; ++c) { e[6 + c] = f16[(size_t)c * NPT + j2] - c16[c]; e[22 + c] = c16[c]; }
#pragma unroll
        for (int o = 0; o < 32; ++o) {
            float y = 0.f;
#pragma unroll
            for (int c = 0; c < 38; ++c) y += w[o * 38 + c] * e[c];
            ymax[o] = fmaxf(ymax[o], y);
            s1[o] += y; s2[o] += y * y;
        }
    }
#pragma unroll
    for (int o = 0; o < 32; ++o) pre[((size_t)b * 32 + o) * NPT + n] = ymax[o];

    __shared__ float red[8 * 64];
    const int lane = tid & 31, wv = tid >> 5;
#pragma unroll
    for (int o = 0; o < 32; ++o) {
        float a = s1[o], q = s2[o];
        for (int off = 16; off > 0; off >>= 1) { a += __shfl_down(a, off); q += __shfl_down(q, off); }
        if (lane == 0) { red[wv * 64 + o] = a; red[wv * 64 + 32 + o] = q; }
    }
    __syncthreads();
    if (tid < 64) {
        float s = 0.f;
        for (int w2 = 0; w2 < 8; ++w2) s += red[w2 * 64 + tid];
        part[(size_t)(blockIdx.y * gridDim.x + blockIdx.x) * 64 + tid] = s;
    }
}

// ---------------------------------------------------------------------------
// BN finalize: fixed-order sum of 128 block partials -> scale/bias
// ---------------------------------------------------------------------------
__global__ void bn_finalize_kernel(const float* __restrict__ part, int nblocks,
                                   const float* __restrict__ g, const float* __restrict__ bias,
                                   float* __restrict__ ab, int Cout) {
    int o = threadIdx.x;
    if (o < Cout) {
        float s1 = 0.f, s2 = 0.f;
        for (int blk = 0; blk < nblocks; ++blk) {
            s1 += part[(size_t)blk * 64 + o];
            s2 += part[(size_t)blk * 64 + 32 + o];
        }
        float m = s1 / BN_CNT;
        float v = s2 / BN_CNT - m * m;
        float a = g[o] * rsqrtf(v + 1e-5f);
        ab[o]        = a;
        ab[Cout + o] = bias[o] - m * a;
    }
}

__global__ void bn_apply_kernel(const float* __restrict__ pre, const float* __restrict__ ab,
                                float* __restrict__ out, int Cout, int total) {
    int i = blockIdx.x * 256 + threadIdx.x;
    if (i < total) {
        int o = (i / NPT) % Cout;
        out[i] = lrelu(ab[o] * pre[i] + ab[Cout + o]);
    }
}

// ---------------------------------------------------------------------------
// h[b][j] = mean over 64 channels (xd1:0-31, xd2:32-63) of points n%64==j
// ---------------------------------------------------------------------------
__global__ void h_kernel(const float* __restrict__ xd1, const float* __restrict__ xd2,
                         float* __restrict__ h) {
    int b = blockIdx.x, j = threadIdx.x;
    float s = 0.f;
    for (int c = 0; c < 32; ++c)
        for (int n = j; n < NPT; n += 64) s += xd1[((size_t)b * 32 + c) * NPT + n];
    for (int c = 0; c < 32; ++c)
        for (int n = j; n < NPT; n += 64) s += xd2[((size_t)b * 32 + c) * NPT + n];
    h[b * 64 + j] = s * (1.0f / 4096.0f);
}

// ---------------------------------------------------------------------------
// MLP head: 64 -> 64 -> 32 -> 11 with lrelu, B=8. One block.
// ---------------------------------------------------------------------------
__global__ __launch_bounds__(512) void head_kernel(const float* __restrict__ h,
                                                   const float* __restrict__ w1, const float* __restrict__ b1,
                                                   const float* __restrict__ w2, const float* __restrict__ b2,
                                                   const float* __restrict__ w3, const float* __restrict__ b3,
                                                   float* __restrict__ out) {
    __shared__ float h1[NB * 64];
    __shared__ float h2[NB * 32];
    const int t = threadIdx.x;
    if (t < NB * 64) {
        int b = t / 64, o = t % 64;
        float s = b1[o];
        for (int c = 0; c < 64; ++c) s += w1[o * 64 + c] * h[b * 64 + c];
        h1[t] = lrelu(s);
    }
    __syncthreads();
    if (t < NB * 32) {
        int b = t / 32, o = t % 32;
        float s = b2[o];
        for (int c = 0; c < 64; ++c) s += w2[o * 64 + c] * h1[b * 64 + c];
        h2[t] = lrelu(s);
    }
    __syncthreads();
    if (t < NB * 11) {
        int b = t / 11, o = t % 11;
        float s = b3[o];
        for (int c = 0; c < 32; ++c) s += w3[o * 32 + c] * h2[b * 32 + c];
        out[t] = lrelu(s);
    }
}

// ---------------------------------------------------------------------------
// workspace layout (float offsets)
// ---------------------------------------------------------------------------
static const size_t OFF_XT    = 0;          // 98304
static const size_t OFF_XX    = 98304;      // 32768
static const size_t OFF_IDX1  = 131072;     // 655360 (ints)
static const size_t OFF_IDX2  = 786432;     // 655360 (ints)
static const size_t OFF_IDX3  = 1441792;    // 655360 (ints)
static const size_t OFF_X1PRE = 2097152;    // 524288
static const size_t OFF_X1    = 2621440;    // 524288
static const size_t OFF_X2PRE = 3145728;    // 1048576
static const size_t OFF_X2    = 4194304;    // 1048576
static const size_t OFF_D1PRE = 5242880;    // 1048576
static const size_t OFF_XD1   = 6291456;    // 1048576
static const size_t OFF_D2PRE = 7340032;    // 1048576
static const size_t OFF_XD2   = 8388608;    // 1048576
static const size_t OFF_PART  = 9437184;    // 8192
static const size_t OFF_AB    = 9445376;    // 256 (4 stages x 64)
static const size_t OFF_H     = 9445632;    // 512

extern "C" void kernel_launch(void* const* d_in, const int* in_sizes, int n_in,
                              void* d_out, int out_size, void* d_ws, size_t ws_size,
                              hipStream_t stream) {
    (void)in_sizes; (void)n_in; (void)out_size; (void)ws_size;
    const float* x        = (const float*)d_in[0];
    const float* conv1_w  = (const float*)d_in[1];
    const float* bn1_g    = (const float*)d_in[2];
    const float* bn1_b    = (const float*)d_in[3];
    const float* conv2_w  = (const float*)d_in[4];
    const float* bn2_g    = (const float*)d_in[5];
    const float* bn2_b    = (const float*)d_in[6];
    const float* convd1_w = (const float*)d_in[7];
    const float* bnd1_g   = (const float*)d_in[8];
    const float* bnd1_b   = (const float*)d_in[9];
    const float* convd2_w = (const float*)d_in[10];
    const float* bnd2_g   = (const float*)d_in[11];
    const float* bnd2_b   = (const float*)d_in[12];
    const float* l1_w     = (const float*)d_in[13];
    const float* l1_b     = (const float*)d_in[14];
    const float* l2_w     = (const float*)d_in[15];
    const float* l2_b     = (const float*)d_in[16];
    const float* l3_w     = (const float*)d_in[17];
    const float* l3_b     = (const float*)d_in[18];

    float* ws = (float*)d_ws;
    float* xt   = ws + OFF_XT;
    float* xx   = ws + OFF_XX;
    int*   idx1 = (int*)(ws + OFF_IDX1);
    int*   idx2 = (int*)(ws + OFF_IDX2);
    int*   idx3 = (int*)(ws + OFF_IDX3);
    float* part = ws + OFF_PART;
    float* ab1  = ws + OFF_AB + 0;
    float* ab2  = ws + OFF_AB + 64;
    float* ab3  = ws + OFF_AB + 128;
    float* ab4  = ws + OFF_AB + 192;

    const dim3 knn_grid(NPT / 16, NB);
    const dim3 conv_grid(NPT / 256, NB);
    const int nblk = (NPT / 256) * NB;   // 128

    // stage 0: layout
    transpose_x_kernel<<<(3 * NB * NPT + 255) / 256, 256, 0, stream>>>(x, xt);

    // stage 1: knn(x, C=3) -> edgeconv 3->16 -> BN/lrelu/max -> x1
    xx_kernel<3><<<(NB * NPT + 255) / 256, 256, 0, stream>>>(xt, xx);
    knn_kernel<3><<<knn_grid, 128, 0, stream>>>(xt, xx, idx1);
    edgeconv_kernel<3, 16><<<conv_grid, 256, 0, stream>>>(xt, idx1, conv1_w, ws + OFF_X1PRE, part);
    bn_finalize_kernel<<<1, 64, 0, stream>>>(part, nblk, bn1_g, bn1_b, ab1, 16);
    bn_apply_kernel<<<(NB * 16 * NPT + 255) / 256, 256, 0, stream>>>(ws + OFF_X1PRE, ab1, ws + OFF_X1, 16, NB * 16 * NPT);

    // stage 2: knn(x1, C=16) -> conv2 (38->32) -> x2 ; and convd1 (gf16 32->32) -> xd1
    xx_kernel<16><<<(NB * NPT + 255) / 256, 256, 0, stream>>>(ws + OFF_X1, xx);
    knn_kernel<16><<<knn_grid, 128, 0, stream>>>(ws + OFF_X1, xx, idx2);
    conv2_kernel<<<conv_grid, 256, 0, stream>>>(xt, idx1, ws + OFF_X1, idx2, conv2_w, ws + OFF_X2PRE, part);
    bn_finalize_kernel<<<1, 64, 0, stream>>>(part, nblk, bn2_g, bn2_b, ab2, 32);
    bn_apply_kernel<<<(NB * 32 * NPT + 255) / 256, 256, 0, stream>>>(ws + OFF_X2PRE, ab2, ws + OFF_X2, 32, NB * 32 * NPT);

    edgeconv_kernel<16, 32><<<conv_grid, 256, 0, stream>>>(ws + OFF_X1, idx2, convd1_w, ws + OFF_D1PRE, part);
    bn_finalize_kernel<<<1, 64, 0, stream>>>(part, nblk, bnd1_g, bnd1_b, ab3, 32);
    bn_apply_kernel<<<(NB * 32 * NPT + 255) / 256, 256, 0, stream>>>(ws + OFF_D1PRE, ab3, ws + OFF_XD1, 32, NB * 32 * NPT);

    // stage 3: knn(x2, C=32) -> convd2 (64->32) -> xd2
    xx_kernel<32><<<(NB * NPT + 255) / 256, 256, 0, stream>>>(ws + OFF_X2, xx);
    knn_kernel<32><<<knn_grid, 128, 0, stream>>>(ws + OFF_X2, xx, idx3);
    edgeconv_kernel<32, 32><<<conv_grid, 256, 0, stream>>>(ws + OFF_X2, idx3, convd2_w, ws + OFF_D2PRE, part);
    bn_finalize_kernel<<<1, 64, 0, stream>>>(part, nblk, bnd2_g, bnd2_b, ab4, 32);
    bn_apply_kernel<<<(NB * 32 * NPT + 255) / 256, 256, 0, stream>>>(ws + OFF_D2PRE, ab4, ws + OFF_XD2, 32, NB * 32 * NPT);

    // head
    h_kernel<<<NB, 64, 0, stream>>>(ws + OFF_XD1, ws + OFF_XD2, ws + OFF_H);
    head_kernel<<<1, 512, 0, stream>>>(ws + OFF_H, l1_w, l1_b, l2_w, l2_b, l3_w, l3_b, (float*)d_out);
}